// RoIPooling_48679159333682
// MI455X (gfx1250) — compile-verified
//
#include <hip/hip_runtime.h>

// Problem constants (from reference setup_inputs)
#define POOL_H 7
#define POOL_W 7
#define FB 8
#define FN 300
#define FH 64
#define FW 64
#define FC 512

typedef float v4f  __attribute__((ext_vector_type(4)));
typedef unsigned int u32x4 __attribute__((ext_vector_type(4)));
typedef int  i32x4 __attribute__((ext_vector_type(4)));
typedef int  i32x8 __attribute__((ext_vector_type(8)));

#if __has_builtin(__builtin_amdgcn_tensor_load_to_lds) && __has_builtin(__builtin_amdgcn_s_wait_tensorcnt)
#define USE_TDM 1
#else
#define USE_TDM 0
#endif

// k/6 for k=0..6, exactly rounded (bit-identical to IEEE f32 division k/6.0f),
// replaces the v_div_scale/v_rcp divide chain with a tiny LUT.
__device__ __constant__ float kT6[7] = {
    0.0f, 0.16666667f, 0.33333334f, 0.5f, 0.6666667f, 0.83333331f, 1.0f
};

// Grid (7, 7, 2400): px = blockIdx.x, py = blockIdx.y, roi = blockIdx.z.
// 128 threads * float4 = 512 channels per pooled cell.
__global__ __launch_bounds__(128) void roi_align_tdm_kernel(
    const float* __restrict__ fmap,   // (8,64,64,512) f32
    const float* __restrict__ rois,   // (8,300,4)     f32
    float* __restrict__ out)          // (8,300,7,7,512) f32
{
    const int px  = blockIdx.x;
    const int py  = blockIdx.y;
    const int roi = blockIdx.z;
    const int b   = roi / FN;                      // single magic-multiply divide
    const int cell = (roi * (POOL_H * POOL_W)) + py * POOL_W + px;

    // reference: boxes = roi[..., [1,0,3,2]] -> y1=roi[1], x1=roi[0], y2=roi[3], x2=roi[2]
    const float bx1 = rois[roi * 4 + 0];
    const float by1 = rois[roi * 4 + 1];
    const float bx2 = rois[roi * 4 + 2];
    const float by2 = rois[roi * 4 + 3];

    const float in_y = (by1 + kT6[py] * (by2 - by1)) * (float)(FH - 1);
    const float in_x = (bx1 + kT6[px] * (bx2 - bx1)) * (float)(FW - 1);
    const bool valid = (in_y >= 0.0f) && (in_y <= (float)(FH - 1)) &&
                       (in_x >= 0.0f) && (in_x <= (float)(FW - 1));

    const float y0f = fminf(fmaxf(floorf(in_y), 0.0f), (float)(FH - 1));
    const float x0f = fminf(fmaxf(floorf(in_x), 0.0f), (float)(FW - 1));
    const float yl = in_y - y0f;
    const float xl = in_x - x0f;
    const int y0 = (int)y0f;
    const int x0 = (int)x0f;
    const int y1 = min(y0 + 1, FH - 1);
    const int x1 = min(x0 + 1, FW - 1);

    const int t = threadIdx.x;     // 0..127
    const int c = t * 4;           // channel offset (float4 per thread)

    v4f v00, v01, v10, v11;

#if USE_TDM
    // Stage the 4 corner rows (4 x 512 f32 = 8 KB) into LDS with one TDM
    // gather-mode descriptor: rows indexed into the flattened (B*H*W, C) tensor.
    __shared__ __align__(16) float tile[4 * FC];

    if (t < 32) {   // single wave builds the D# and issues the tensor DMA
        const unsigned r00 = (unsigned)((b * FH + y0) * FW + x0);
        const unsigned r01 = (unsigned)((b * FH + y0) * FW + x1);
        const unsigned r10 = (unsigned)((b * FH + y1) * FW + x0);
        const unsigned r11 = (unsigned)((b * FH + y1) * FW + x1);

        const unsigned long long gaddr = (unsigned long long)(const void*)fmap;
        const unsigned lds_addr = (unsigned)(unsigned long long)(const void*)&tile[0];

        // ---- D# group 0 (ISA 8.3): count=1, gather_mode=1 (16-bit idx),
        //      lds_addr, global_addr (57b), type=2 ("image")
        u32x4 g0;
        g0[0] = 1u | (1u << 31);                                   // count=1 | gather_mode
        g0[1] = lds_addr;                                          // LDS byte address
        g0[2] = (unsigned)(gaddr & 0xffffffffull);                 // global_addr[31:0]
        g0[3] = (unsigned)((gaddr >> 32) & 0x01ffffffull)          // global_addr[56:32]
              | (2u << 30);                                        // type = 2
        // ---- D# group 1 (ISA 8.4)
        i32x8 g1;
        g1[0] = (2 << 16);                        // workgroup_mask=0, data_size=2 (4 bytes)
        g1[1] = (int)((unsigned)FC << 16);        // tensor_dim0[15:0]=512 (bits 79:48)
        g1[2] = (int)((unsigned)(FB*FH*FW) << 16);// tensor_dim1[15:0]=32768 (#rows, OOB bound)
        g1[3] = (int)((unsigned)FC << 16);        // tile_dim0=512 (bits 127:112)
        g1[4] = 4;                                // tile_dim1 = #valid gather indices
        g1[5] = FC;                               // tensor_dim0_stride = 512 elements
        g1[6] = 0;
        g1[7] = 0;
        // ---- D# groups 2/3 (gather mode, 16-bit row indices; ISA 8.7)
        i32x4 g2;
        g2[0] = (int)(r00 | (r01 << 16));
        g2[1] = (int)(r10 | (r11 << 16));
        g2[2] = 0;
        g2[3] = 0;
        i32x4 g3 = {0, 0, 0, 0};
        i32x8 gpad = {0, 0, 0, 0, 0, 0, 0, 0};    // extra operand in 6-arg builtin form

        __builtin_amdgcn_tensor_load_to_lds(g0, g1, g2, g3, gpad, 0);
        __builtin_amdgcn_s_wait_tensorcnt(0);
    }
    __syncthreads();

    v00 = *(const v4f*)&tile[0 * FC + c];
    v01 = *(const v4f*)&tile[1 * FC + c];
    v10 = *(const v4f*)&tile[2 * FC + c];
    v11 = *(const v4f*)&tile[3 * FC + c];
#else
    // Fallback: direct coalesced b128 gathers (also used by the host pass).
    const size_t base = (size_t)b * (FH * FW * FC);
    v00 = *(const v4f*)(fmap + base + (size_t)(y0 * FW + x0) * FC + c);
    v01 = *(const v4f*)(fmap + base + (size_t)(y0 * FW + x1) * FC + c);
    v10 = *(const v4f*)(fmap + base + (size_t)(y1 * FW + x0) * FC + c);
    v11 = *(const v4f*)(fmap + base + (size_t)(y1 * FW + x1) * FC + c);
#endif

    // Bilinear interpolation, same op order as reference
    v4f top = v00 + (v01 - v00) * xl;
    v4f bot = v10 + (v11 - v10) * xl;
    v4f r   = top + (bot - top) * yl;
    if (!valid) r = v4f{0.0f, 0.0f, 0.0f, 0.0f};

    // Output is a 240MB write-once stream: non-temporal so it does not evict
    // the L2-resident (64MB) feature map.
    __builtin_nontemporal_store(r, (v4f*)(out + (size_t)cell * FC + c));
}

extern "C" void kernel_launch(void* const* d_in, const int* in_sizes, int n_in,
                              void* d_out, int out_size, void* d_ws, size_t ws_size,
                              hipStream_t stream) {
    const float* fmap = (const float*)d_in[0];   // (8,64,64,512) f32
    const float* rois = (const float*)d_in[1];   // (8,300,4)     f32
    float* out = (float*)d_out;                  // (8,300,7,7,512) f32

    dim3 grid(POOL_W, POOL_H, FB * FN);          // (7, 7, 2400)
    roi_align_tdm_kernel<<<grid, dim3(128), 0, stream>>>(fmap, rois, out);
}